// GIN_54065048323041
// MI455X (gfx1250) — compile-verified
//
#include <hip/hip_runtime.h>

// GIN (2-layer, D=128) for gfx1250 / MI455X.
// Pipeline:
//   z0 = x; z0[dst] += x[src]           (copy + scatter, native f32 L2 atomics)
//   h  = relu(mlp1(z0)); z1 = h         (fused WMMA MLP, dual store)
//   z1[dst] += h[src]                   (scatter)
//   out = mlp2(z1)                      (fused WMMA MLP)

typedef __attribute__((ext_vector_type(2))) float v2f;
typedef __attribute__((ext_vector_type(8))) float v8f;

#define D 128
#define LDS_STRIDE 132   // 128 + 4 pad: conflict-free ds access patterns

// ---------------------------------------------------------------------------
// Streaming copy: out = in  (float4 granularity)
// ---------------------------------------------------------------------------
__global__ __launch_bounds__(256) void gin_copy(const float* __restrict__ in,
                                                float* __restrict__ out, int n4) {
    int i = blockIdx.x * 256 + threadIdx.x;
    if (i < n4) ((float4*)out)[i] = ((const float4*)in)[i];
}

// ---------------------------------------------------------------------------
// Edge scatter-add: acc[dst[e]][:] += h[src[e]][:]
// One wave32 per edge; each lane handles one float4 -> 4 native f32 atomics
// (global_atomic_add_f32, device scope, immediate offsets off one addr pair).
// Edge indices are wave-uniform -> readfirstlane so they become s_loads.
// ---------------------------------------------------------------------------
#define GIN_ATOMIC_ADD_F32(PTR, OFFSTR, VAL)                                  \
    asm volatile("global_atomic_add_f32 %0, %1, off offset:" OFFSTR           \
                 " scope:SCOPE_DEV"                                           \
                 :: "v"(PTR), "v"(VAL) : "memory")

__global__ __launch_bounds__(256) void gin_scatter(const float* __restrict__ H,
                                                   const long long* __restrict__ src,
                                                   const long long* __restrict__ dst,
                                                   float* __restrict__ acc, int nEdges) {
    int gid  = blockIdx.x * 256 + threadIdx.x;
    int lane = gid & 31;
    int e    = __builtin_amdgcn_readfirstlane(gid >> 5);   // wave-uniform edge id
    if (e >= nEdges) return;
    long long s = src[e];
    long long d = dst[e];
    const float4 v = *(const float4*)(H + s * D + lane * 4);
    float* p = acc + d * D + lane * 4;
    GIN_ATOMIC_ADD_F32(p, "0",  v.x);
    GIN_ATOMIC_ADD_F32(p, "4",  v.y);
    GIN_ATOMIC_ADD_F32(p, "8",  v.z);
    GIN_ATOMIC_ADD_F32(p, "12", v.w);
}

// ---------------------------------------------------------------------------
// Fused MLP:  Out = (relu_out? relu:id)( relu(Z@Wa + ba) @ Wb + bb )
// Optionally dual-stores the result to Out2 (feeds the next layer's scatter
// accumulator without a separate copy pass).
//
// 4 waves / block, each wave owns a private 16x128 row strip in LDS.
// A fragments (16x4 f32) via ds_load, B fragments (4x16 f32) from the
// L2-resident 64KB weights. All 8 N-tile accumulators live in registers so
// the LDS tile can be overwritten in place between the two GEMMs
// (wave-private tile; DS ops are in-order per wave).
//
// WMMA f32 16x16x4 operand mapping (ISA 7.12.2):
//   A: lanes 0-15 -> M=lane, K=0..1; lanes 16-31 -> M=lane-16, K=2..3
//   B: lanes 0-15 -> K=0..1, N=lane; lanes 16-31 -> K=2..3, N=lane-16
//   C: vgpr r: lanes 0-15 -> (M=r, N=lane); lanes 16-31 -> (M=r+8, N=lane-16)
// ---------------------------------------------------------------------------
__global__ __launch_bounds__(128) void gin_mlp_wmma(const float* __restrict__ Z,
                                                    const float* __restrict__ Wa,
                                                    const float* __restrict__ ba,
                                                    const float* __restrict__ Wb,
                                                    const float* __restrict__ bb,
                                                    float* __restrict__ Out,
                                                    float* __restrict__ Out2,
                                                    int n, int relu_out) {
    __shared__ float sZ[4][16][LDS_STRIDE];

    const int wave    = threadIdx.x >> 5;
    const int lane    = threadIdx.x & 31;
    const int half    = lane >> 4;      // 0: lanes 0-15, 1: lanes 16-31
    const int l15     = lane & 15;
    const int rowBase = blockIdx.x * 64 + wave * 16;

    // --- Stage 16x128 input tile into the wave-private LDS tile (coalesced).
#pragma unroll 4
    for (int r = 0; r < 16; ++r) {
        int row = rowBase + r;
        float4 v = make_float4(0.f, 0.f, 0.f, 0.f);
        if (row < n) v = *(const float4*)(Z + (size_t)row * D + lane * 4);
        *(float4*)(&sZ[wave][r][lane * 4]) = v;
    }

    v8f acc[8];

    // ================= GEMM 1: z' = relu(Z @ Wa + ba) =================
#pragma unroll
    for (int nt = 0; nt < 8; ++nt) acc[nt] = (v8f){};

#pragma unroll 4
    for (int k = 0; k < 32; ++k) {
        const int krow = 4 * k + 2 * half;
        v2f a = *(const v2f*)(&sZ[wave][l15][krow]);
        const float* wp = Wa + krow * D + l15;
#pragma unroll
        for (int nt = 0; nt < 8; ++nt) {
            v2f b;
            b.x = wp[nt * 16];
            b.y = wp[D + nt * 16];
            acc[nt] = __builtin_amdgcn_wmma_f32_16x16x4_f32(
                false, a, false, b, (short)0, acc[nt], false, false);
        }
    }

    // bias + relu, write hidden activation back into the wave-private tile
#pragma unroll
    for (int nt = 0; nt < 8; ++nt) {
        const int col = nt * 16 + l15;
        const float bias = ba[col];
#pragma unroll
        for (int r = 0; r < 8; ++r) {
            float v = acc[nt][r] + bias;
            v = fmaxf(v, 0.0f);
            sZ[wave][r + 8 * half][col] = v;
        }
    }

    // ================= GEMM 2: out = z' @ Wb + bb =================
#pragma unroll
    for (int nt = 0; nt < 8; ++nt) acc[nt] = (v8f){};

#pragma unroll 4
    for (int k = 0; k < 32; ++k) {
        const int krow = 4 * k + 2 * half;
        v2f a = *(const v2f*)(&sZ[wave][l15][krow]);
        const float* wp = Wb + krow * D + l15;
#pragma unroll
        for (int nt = 0; nt < 8; ++nt) {
            v2f b;
            b.x = wp[nt * 16];
            b.y = wp[D + nt * 16];
            acc[nt] = __builtin_amdgcn_wmma_f32_16x16x4_f32(
                false, a, false, b, (short)0, acc[nt], false, false);
        }
    }

    // bias (+ optional relu) and guarded global store(s)
#pragma unroll
    for (int nt = 0; nt < 8; ++nt) {
        const int col = nt * 16 + l15;
        const float bias = bb[col];
#pragma unroll
        for (int r = 0; r < 8; ++r) {
            const int row = rowBase + r + 8 * half;
            float v = acc[nt][r] + bias;
            if (relu_out) v = fmaxf(v, 0.0f);
            if (row < n) {
                Out[(size_t)row * D + col] = v;
                if (Out2) Out2[(size_t)row * D + col] = v;
            }
        }
    }
}

// ---------------------------------------------------------------------------
// Host-side launcher
// ---------------------------------------------------------------------------
extern "C" void kernel_launch(void* const* d_in, const int* in_sizes, int n_in,
                              void* d_out, int out_size, void* d_ws, size_t ws_size,
                              hipStream_t stream) {
    const float*     x   = (const float*)d_in[0];
    const long long* ei  = (const long long*)d_in[1];   // [2, E] int64
    const float*     W1a = (const float*)d_in[2];
    const float*     b1a = (const float*)d_in[3];
    const float*     W1b = (const float*)d_in[4];
    const float*     b1b = (const float*)d_in[5];
    const float*     W2a = (const float*)d_in[6];
    const float*     b2a = (const float*)d_in[7];
    const float*     W2b = (const float*)d_in[8];
    const float*     b2b = (const float*)d_in[9];

    const int nNodes = in_sizes[0] / D;   // 50000
    const int nEdges = in_sizes[1] / 2;   // 600000
    const long long* srcI = ei;
    const long long* dstI = ei + nEdges;

    float* z0 = (float*)d_ws;
    float* h  = z0 + (size_t)nNodes * D;
    float* z1 = h  + (size_t)nNodes * D;
    float* out = (float*)d_out;

    const int n4 = nNodes * D / 4;
    dim3 cpyB(256), cpyG((n4 + 255) / 256);
    dim3 scB(256),  scG(((size_t)nEdges * 32 + 255) / 256);
    dim3 mlpB(128), mlpG((nNodes + 63) / 64);

    // ---- Layer 1 ----
    gin_copy    <<<cpyG, cpyB, 0, stream>>>(x, z0, n4);
    gin_scatter <<<scG,  scB,  0, stream>>>(x, srcI, dstI, z0, nEdges);
    gin_mlp_wmma<<<mlpG, mlpB, 0, stream>>>(z0, W1a, b1a, W1b, b1b, h, z1, nNodes, 1);

    // ---- Layer 2 ----
    gin_scatter <<<scG,  scB,  0, stream>>>(h, srcI, dstI, z1, nEdges);
    gin_mlp_wmma<<<mlpG, mlpB, 0, stream>>>(z1, W2a, b2a, W2b, b2b, out, nullptr, nNodes, 0);
}